// MicroVoxelSpatialEncoder_5540507812266
// MI455X (gfx1250) — compile-verified
//
#include <hip/hip_runtime.h>
#include <hip/hip_bf16.h>
#include <math.h>

typedef __attribute__((ext_vector_type(2))) float v2f;
typedef __attribute__((ext_vector_type(8))) float v8f;

#define BATCH 2
#define NPTS  16384
#define BN    (BATCH * NPTS)      // 32768
#define FINC  64
#define DIMC  128
#define NXV   10
#define NYV   10
#define NTV   200
#define MVOX  (NXV * NYV * NTV)   // 20000
#define BM    (BATCH * MVOX)      // 40000
#define KNBR  27

// ---------------------------------------------------------------------------
// Row-major GEMM: C[Mrows x 128] = A[Mrows x K] @ W[K x 128] (+bias).
// K compile-time (64/128). Block = 8 waves; each wave computes TWO 16x16
// output tiles (cols col0+0..15 and col0+16..31) sharing each A fragment, so
// one global_load_b64 feeds two v_wmma_f32_16x16x4_f32. The block's 32-col W
// panel is staged transposed in LDS (pitch K+4 -> conflict-free ds_load_b64).
// ---------------------------------------------------------------------------
template <int K, bool HASBIAS>
__global__ __launch_bounds__(256) void gemm_wmma_kernel(
    const float* __restrict__ A, const float* __restrict__ W,
    const float* __restrict__ bias, float* __restrict__ C, int Mrows) {
  constexpr int NCOLS = DIMC;
  constexpr int KPAD  = K + 4;
  __shared__ float panelT[32 * KPAD];
  const int tid  = threadIdx.x;
  const int col0 = blockIdx.y * 32;
  for (int idx = tid; idx < 32 * K; idx += 256) {
    const int k = idx >> 5, n = idx & 31;
    panelT[((n >> 4) * 16 + (n & 15)) * KPAD + k] = W[k * NCOLS + col0 + n];
  }
  __syncthreads();

  const int lane = tid & 31;
  const int wv   = tid >> 5;
  const int row0 = (blockIdx.x * 8 + wv) * 16;
  if (row0 >= Mrows) return;                    // wave-uniform
  const int m  = lane & 15;
  const int hi = lane >> 4;
  const int hk = hi << 1;                       // K sub-offset 0 or 2
  const float* ap  = A + (size_t)(row0 + m) * K + hk;  // A: M=lane row
  const float* bp0 = panelT + m * KPAD + hk;           // B: N=lane col, panel 0
  const float* bp1 = bp0 + 16 * KPAD;                  // panel 1 (cols +16)
  v8f c0 = {}, c1 = {};
#pragma unroll 8
  for (int kk = 0; kk < K; kk += 4) {
    const v2f a  = *(const v2f*)ap;             // global_load_b64
    const v2f b0 = *(const v2f*)bp0;            // ds_load_b64
    const v2f b1 = *(const v2f*)bp1;            // ds_load_b64
    ap += 4;
    bp0 += 4;
    bp1 += 4;
    c0 = __builtin_amdgcn_wmma_f32_16x16x4_f32(false, a, false, b0,
                                               (short)0, c0, false, false);
    c1 = __builtin_amdgcn_wmma_f32_16x16x4_f32(false, a, false, b1,
                                               (short)0, c1, false, false);
  }
  const int colg = col0 + m;
  const float bv0 = HASBIAS ? bias[colg] : 0.0f;
  const float bv1 = HASBIAS ? bias[colg + 16] : 0.0f;
  float* cp = C + (size_t)(row0 + hi * 8) * NCOLS + colg;
#pragma unroll
  for (int r = 0; r < 8; ++r) {                 // C/D layout: VGPR r -> M=r / r+8
    cp[r * NCOLS]      = c0[r] + bv0;
    cp[r * NCOLS + 16] = c1[r] + bv1;
  }
}

// Copy pos_emb (27x128) into a zero-padded 32x128 buffer (GEMM needs no guards).
__global__ __launch_bounds__(256) void pad_pos_kernel(
    const float* __restrict__ pos, float* __restrict__ dst) {
  const int i = blockIdx.x * 256 + threadIdx.x;
  if (i >= 32 * DIMC) return;
  dst[i] = (i < KNBR * DIMC) ? pos[i] : 0.0f;
}

// ---------------------------------------------------------------------------
// Voxelize + scatter-sum. One wave per point.
// ---------------------------------------------------------------------------
__global__ __launch_bounds__(256) void voxel_scatter_kernel(
    const float* __restrict__ coords, const float* __restrict__ feat,
    float* __restrict__ pooled, float* __restrict__ cnt,
    int4* __restrict__ vidx) {
  const int gw = (blockIdx.x * 256 + threadIdx.x) >> 5;
  const int lane = threadIdx.x & 31;
  if (gw >= BN) return;
  const float cx = coords[gw * 3 + 0];
  const float cy = coords[gw * 3 + 1];
  const float ct = coords[gw * 3 + 2];
  const int ix = (int)(fminf(fmaxf(cx, 0.0f), 1.0f) * (float)(NXV - 1));
  const int iy = (int)(fminf(fmaxf(cy, 0.0f), 1.0f) * (float)(NYV - 1));
  const int it = (int)(fminf(fmaxf(ct, 0.0f), 1.0f) * (float)(NTV - 1));
  const int b  = gw >> 14;                      // NPTS = 2^14
  const int gid = ix + iy * NXV + it * (NXV * NYV) + b * MVOX;
  if (lane == 0) {
    vidx[gw] = make_int4(ix, iy, it, gid);
    atomicAdd(&cnt[gid], 1.0f);
  }
  const float4 fc = ((const float4*)(feat + (size_t)gw * DIMC))[lane];
  float* base = pooled + (size_t)gid * DIMC + lane * 4;
  atomicAdd(base + 0, fc.x);
  atomicAdd(base + 1, fc.y);
  atomicAdd(base + 2, fc.z);
  atomicAdd(base + 3, fc.w);
}

// In-place scatter-mean: pooled /= max(cnt,1)  (pooled becomes `table`)
__global__ __launch_bounds__(256) void normalize_kernel(
    float* __restrict__ pooled, const float* __restrict__ cnt) {
  const int i = blockIdx.x * 256 + threadIdx.x;
  if (i >= BM * DIMC) return;
  pooled[i] *= 1.0f / fmaxf(cnt[i >> 7], 1.0f);
}

// ---------------------------------------------------------------------------
// Attention: one wave per point, online softmax over 27 neighbor voxels.
// tableK/tableV (20.5 MB each) stay L2-resident; each neighbor row is a
// 512-byte coalesced gather (lane L reads float4 chunk L). Per-axis bounds
// rejection skips 9 (or 3) neighbors with a single compare.
// ---------------------------------------------------------------------------
__global__ __launch_bounds__(256) void attn_kernel(
    const float* __restrict__ q,  const float* __restrict__ tK,
    const float* __restrict__ tV, const float* __restrict__ pK,
    const float* __restrict__ pV, const int4* __restrict__ vidx,
    float* __restrict__ o) {
  const int gw = (blockIdx.x * 256 + threadIdx.x) >> 5;
  const int lane = threadIdx.x & 31;
  if (gw >= BN) return;
  const int b = gw >> 14;
  const int4 vi = vidx[gw];
  const float4 qc = ((const float4*)(q + (size_t)gw * DIMC))[lane];
  float mrun = -3.0e38f, lsum = 0.0f;
  float4 oc = make_float4(0.0f, 0.0f, 0.0f, 0.0f);
  int j = 0;                                     // pos_emb row (dx fastest)
  for (int dt = -1; dt <= 1; ++dt) {
    const int nt = vi.z + dt;
    if (nt < 0 || nt >= NTV) { j += 9; continue; }   // wave-uniform
    for (int dy = -1; dy <= 1; ++dy) {
      const int ny = vi.y + dy;
      if (ny < 0 || ny >= NYV) { j += 3; continue; }
      for (int dx = -1; dx <= 1; ++dx, ++j) {
        const int nx = vi.x + dx;
        if (nx < 0 || nx >= NXV) continue;
        const int sel = nx + ny * NXV + nt * (NXV * NYV) + b * MVOX;
        const float4 kp = ((const float4*)(pK + (size_t)j * DIMC))[lane];
        const float4 kr = ((const float4*)(tK + (size_t)sel * DIMC))[lane];
        float s = (kr.x + kp.x) * qc.x + (kr.y + kp.y) * qc.y +
                  (kr.z + kp.z) * qc.z + (kr.w + kp.w) * qc.w;
        s += __shfl_xor(s, 16, 32);
        s += __shfl_xor(s, 8, 32);
        s += __shfl_xor(s, 4, 32);
        s += __shfl_xor(s, 2, 32);
        s += __shfl_xor(s, 1, 32);
        s *= 0.08838834764831845f;               // 1/sqrt(128)
        const float4 vp = ((const float4*)(pV + (size_t)j * DIMC))[lane];
        const float4 vr = ((const float4*)(tV + (size_t)sel * DIMC))[lane];
        const float mn = fmaxf(mrun, s);
        const float cs = __expf(mrun - mn);      // 0 on first valid neighbor
        const float w  = __expf(s - mn);
        lsum = lsum * cs + w;
        oc.x = oc.x * cs + w * (vr.x + vp.x);
        oc.y = oc.y * cs + w * (vr.y + vp.y);
        oc.z = oc.z * cs + w * (vr.z + vp.z);
        oc.w = oc.w * cs + w * (vr.w + vp.w);
        mrun = mn;
      }
    }
  }
  const float inv = 1.0f / lsum;                 // center voxel always valid
  ((float4*)(o + (size_t)gw * DIMC))[lane] =
      make_float4(oc.x * inv, oc.y * inv, oc.z * inv, oc.w * inv);
}

// ---------------------------------------------------------------------------
// h = feat + hraw ; LayerNorm(h) ; mask-select. 16 threads per row, each
// holding 8 elements in registers; 16-lane shfl_xor reduction tree (no LDS).
// ---------------------------------------------------------------------------
__global__ __launch_bounds__(256) void ln_kernel(
    const float* __restrict__ hraw, const float* __restrict__ feat,
    const unsigned char* __restrict__ vmask,
    const float* __restrict__ gamma, const float* __restrict__ beta,
    float* __restrict__ out) {
  const int tid = threadIdx.x;
  const int r   = tid >> 4;                     // 16 rows per block
  const int seg = tid & 15;                     // 16 threads per row
  const int row = blockIdx.x * 16 + r;
  const float4* hp = (const float4*)(hraw + (size_t)row * DIMC);
  const float4* fp = (const float4*)(feat + (size_t)row * DIMC);
  const float4 f0 = fp[seg * 2 + 0];
  const float4 f1 = fp[seg * 2 + 1];
  float4 h0 = hp[seg * 2 + 0];
  float4 h1 = hp[seg * 2 + 1];
  h0.x += f0.x; h0.y += f0.y; h0.z += f0.z; h0.w += f0.w;
  h1.x += f1.x; h1.y += f1.y; h1.z += f1.z; h1.w += f1.w;
  float s  = h0.x + h0.y + h0.z + h0.w + h1.x + h1.y + h1.z + h1.w;
  float q2 = h0.x * h0.x + h0.y * h0.y + h0.z * h0.z + h0.w * h0.w +
             h1.x * h1.x + h1.y * h1.y + h1.z * h1.z + h1.w * h1.w;
  s  += __shfl_xor(s, 8, 32);  q2 += __shfl_xor(q2, 8, 32);
  s  += __shfl_xor(s, 4, 32);  q2 += __shfl_xor(q2, 4, 32);
  s  += __shfl_xor(s, 2, 32);  q2 += __shfl_xor(q2, 2, 32);
  s  += __shfl_xor(s, 1, 32);  q2 += __shfl_xor(q2, 1, 32);
  const float mu  = s * (1.0f / DIMC);
  const float var = q2 * (1.0f / DIMC) - mu * mu;
  const float rs  = __frsqrt_rn(var + 1e-5f);
  const bool ok = vmask[row] != 0;
  float4 o0, o1;
  const int c0 = seg * 8;
  o0.x = ok ? (h0.x - mu) * rs * gamma[c0 + 0] + beta[c0 + 0] : f0.x;
  o0.y = ok ? (h0.y - mu) * rs * gamma[c0 + 1] + beta[c0 + 1] : f0.y;
  o0.z = ok ? (h0.z - mu) * rs * gamma[c0 + 2] + beta[c0 + 2] : f0.z;
  o0.w = ok ? (h0.w - mu) * rs * gamma[c0 + 3] + beta[c0 + 3] : f0.w;
  o1.x = ok ? (h1.x - mu) * rs * gamma[c0 + 4] + beta[c0 + 4] : f1.x;
  o1.y = ok ? (h1.y - mu) * rs * gamma[c0 + 5] + beta[c0 + 5] : f1.y;
  o1.z = ok ? (h1.z - mu) * rs * gamma[c0 + 6] + beta[c0 + 6] : f1.z;
  o1.w = ok ? (h1.w - mu) * rs * gamma[c0 + 7] + beta[c0 + 7] : f1.w;
  float4* op = (float4*)(out + (size_t)row * DIMC);
  op[seg * 2 + 0] = o0;
  op[seg * 2 + 1] = o1;
}

// ---------------------------------------------------------------------------
extern "C" void kernel_launch(void* const* d_in, const int* in_sizes, int n_in,
                              void* d_out, int out_size, void* d_ws,
                              size_t ws_size, hipStream_t stream) {
  const float* features = (const float*)d_in[0];          // (B,N,64)
  const float* coords   = (const float*)d_in[1];          // (B,N,3)
  const unsigned char* vmaskp = (const unsigned char*)d_in[2]; // (B,N) bool
  const float* Wf = (const float*)d_in[3];
  const float* bf = (const float*)d_in[4];
  const float* Wq = (const float*)d_in[5];
  const float* bq = (const float*)d_in[6];
  const float* Wk = (const float*)d_in[7];
  const float* bk = (const float*)d_in[8];
  const float* Wv = (const float*)d_in[9];
  const float* bv = (const float*)d_in[10];
  const float* Wo = (const float*)d_in[11];
  const float* bo = (const float*)d_in[12];
  const float* pos_emb = (const float*)d_in[13];          // (27,128)
  const float* gamma = (const float*)d_in[14];
  const float* beta  = (const float*)d_in[15];
  float* outp = (float*)d_out;

  // ---- workspace carve-up (256B aligned) ----
  char* base = (char*)d_ws;
  size_t off = 0;
  auto carve = [&](size_t bytes) {
    char* p = base + off;
    off += (bytes + 255) & ~(size_t)255;
    return p;
  };
  float* feat   = (float*)carve((size_t)BN * DIMC * 4);   // 16.8 MB
  float* q      = (float*)carve((size_t)BN * DIMC * 4);   // 16.8 MB
  float* o      = (float*)carve((size_t)BN * DIMC * 4);   // 16.8 MB
  float* hraw   = (float*)carve((size_t)BN * DIMC * 4);   // 16.8 MB
  float* pooled = (float*)carve((size_t)BM * DIMC * 4);   // 20.5 MB
  float* cnt    = (float*)carve((size_t)BM * 4);
  float* tabK   = (float*)carve((size_t)BM * DIMC * 4);   // 20.5 MB
  float* tabV   = (float*)carve((size_t)BM * DIMC * 4);   // 20.5 MB
  float* posPad = (float*)carve((size_t)32 * DIMC * 4);
  float* posK   = (float*)carve((size_t)32 * DIMC * 4);
  float* posV   = (float*)carve((size_t)32 * DIMC * 4);
  int4*  vidx   = (int4*)carve((size_t)BN * sizeof(int4));

  // 1) zero the scatter accumulators (required every call: graph replays)
  hipMemsetAsync(pooled, 0, (size_t)BM * DIMC * 4, stream);
  hipMemsetAsync(cnt, 0, (size_t)BM * 4, stream);

  dim3 blk(256);
  const dim3 gBN(BN / 16 / 8, DIMC / 32);                 // (256, 4)
  const dim3 gBM((BM / 16 + 7) / 8, DIMC / 32);           // (313, 4)
  const dim3 gPOS(1, DIMC / 32);                          // (1, 4)

  // 2) feat = features @ Wf + bf       (32768 x 64 x 128)
  gemm_wmma_kernel<FINC, true><<<gBN, blk, 0, stream>>>(features, Wf, bf, feat, BN);
  // 3) voxel scatter-sum
  voxel_scatter_kernel<<<dim3(BN * 32 / 256), blk, 0, stream>>>(
      coords, feat, pooled, cnt, vidx);
  // 4) scatter-mean (pooled -> table, in place)
  normalize_kernel<<<dim3((BM * DIMC + 255) / 256), blk, 0, stream>>>(pooled, cnt);
  // 5) tableK = table @ Wk + bk ; tableV = table @ Wv + bv   (40000 x 128 x 128)
  gemm_wmma_kernel<DIMC, true><<<gBM, blk, 0, stream>>>(pooled, Wk, bk, tabK, BM);
  gemm_wmma_kernel<DIMC, true><<<gBM, blk, 0, stream>>>(pooled, Wv, bv, tabV, BM);
  // 6) q = feat @ Wq + bq              (32768 x 128 x 128)
  gemm_wmma_kernel<DIMC, true><<<gBN, blk, 0, stream>>>(feat, Wq, bq, q, BN);
  // 7) posK = pos_emb @ Wk ; posV = pos_emb @ Wv   (padded to 32 rows, no bias)
  pad_pos_kernel<<<dim3((32 * DIMC + 255) / 256), blk, 0, stream>>>(pos_emb, posPad);
  gemm_wmma_kernel<DIMC, false><<<gPOS, blk, 0, stream>>>(posPad, Wk, nullptr, posK, 32);
  gemm_wmma_kernel<DIMC, false><<<gPOS, blk, 0, stream>>>(posPad, Wv, nullptr, posV, 32);
  // 8) attention (one wave per point, online softmax over 27 L2-resident rows)
  attn_kernel<<<dim3(BN * 32 / 256), blk, 0, stream>>>(
      q, tabK, tabV, posK, posV, vidx, o);
  // 9) hraw = o @ Wo + bo              (32768 x 128 x 128)
  gemm_wmma_kernel<DIMC, true><<<gBN, blk, 0, stream>>>(o, Wo, bo, hraw, BN);
  // 10) residual + LayerNorm + mask-select
  ln_kernel<<<dim3(BN / 16), blk, 0, stream>>>(hraw, feat, vmaskp, gamma, beta, outp);
}